// ChebyshevDescriptor_74904229642925
// MI455X (gfx1250) — compile-verified
//
#include <hip/hip_runtime.h>

typedef float v2f __attribute__((ext_vector_type(2)));
typedef float v8f __attribute__((ext_vector_type(8)));

#define NATOMS   20000
#define KNBR     24
#define RADO     16
#define ANGO     8
#define RAD_CUT  8.0f
#define ANG_CUT  6.5f
#define MIN_CUT  0.55f
#define NOUT     52
#define PI_F     3.14159265358979f
#define RSTR     36            // reduction-scratch row stride (dwords): 16B aligned, bank-spread

// broadcast lane q's value to all lanes via v_readlane (no LDS traffic)
__device__ __forceinline__ float lane_bcast(float v, int q) {
    return __uint_as_float(__builtin_amdgcn_readlane(__float_as_uint(v), q));
}

// Chebyshev T_0..T_8 of clamped cos; accumulate weighted / spin-weighted sums.
__device__ __forceinline__ void cheb_acc(float cosv, float wgt, float ws,
                                         float au[9], float aw[9]) {
    float c = fminf(1.0f, fmaxf(-1.0f, cosv));
    float tm = 1.0f, t = c;
    au[0] += wgt;      aw[0] += ws;
    au[1] += t * wgt;  aw[1] += t * ws;
#pragma unroll
    for (int i = 2; i <= ANGO; i++) {
        float tn = 2.0f * c * t - tm;
        tm = t; t = tn;
        au[i] += t * wgt;  aw[i] += t * ws;
    }
}

__global__ __launch_bounds__(256)
void cheb_desc_kernel(const float* __restrict__ pos,
                      const int*   __restrict__ species,
                      const int*   __restrict__ nbr,
                      float*       __restrict__ out) {
    const int lane = threadIdx.x & 31;
    const int wave = threadIdx.x >> 5;
    const int atom = blockIdx.x * 8 + wave;   // 20000/8 = 2500 blocks, no tail

    __shared__ float4 s_u [8][32];            // unit bond vectors (w unused)
    __shared__ float2 s_ws[8][32];            // {wa (angular weight), sj (typespin)}
    __shared__ __align__(16) float s_red[8][18 * RSTR];  // angular transpose scratch

    // ---------------- per-neighbor gather (lane = neighbor slot) -------------
    const bool isnbr = lane < KNBR;
    const int  j     = nbr[atom * KNBR + (isnbr ? lane : 0)];

    const float pix = pos[3 * atom + 0];
    const float piy = pos[3 * atom + 1];
    const float piz = pos[3 * atom + 2];

    float rx = pos[3 * j + 0] - pix;
    float ry = pos[3 * j + 1] - piy;
    float rz = pos[3 * j + 2] - piz;
    float d   = sqrtf(rx * rx + ry * ry + rz * rz);
    float inv = 1.0f / fmaxf(d, 1e-9f);
    float ux = rx * inv, uy = ry * inv, uz = rz * inv;
    float sj = (float)(2 * species[j] - 1);          // TYPESPIN {-1,+1}
    if (!isnbr) { ux = 0.f; uy = 0.f; uz = 0.f; sj = 0.f; }

    // ---------------- radial features (closed form, readlane broadcast) ------
    // T_c(x) = cos(c*acos(x)); out-of-range x is weight-masked so clamp is exact.
    // Lane c < 17 owns coefficient c for BOTH flavors: one cos feeds both sums.
    {
        const bool  mrad = isnbr && (d <= RAD_CUT) && (d > MIN_CUT);
        const float x    = 2.0f * (d - MIN_CUT) / (RAD_CUT - MIN_CUT) - 1.0f;
        const float ac   = acosf(fminf(1.0f, fmaxf(-1.0f, x)));
        const float fc   = 0.5f * (__cosf(PI_F * d / RAD_CUT) + 1.0f);
        const float wr   = mrad ? fc : 0.0f;
        const float wrs  = wr * sj;

        const float cf = (float)lane;                // coefficient index (lanes >=17 unused)
        float accU = 0.0f, accW = 0.0f;
#pragma unroll
        for (int q = 0; q < KNBR; q++) {
            const float acq  = lane_bcast(ac,  q);
            const float wrq  = lane_bcast(wr,  q);
            const float wrsq = lane_bcast(wrs, q);
            const float t    = __cosf(cf * acq);     // T_lane(x_q)
            accU = fmaf(t, wrq,  accU);
            accW = fmaf(t, wrsq, accW);
        }
        if (lane < 17) {
            out[atom * NOUT + lane]      = accU;     // rad_un[lane]
            out[atom * NOUT + 17 + lane] = accW;     // rad_w[lane]
        }
    }

    // ---------------- angular weights + per-wave LDS tables ------------------
    const bool  mang = isnbr && (d <= ANG_CUT) && (d > MIN_CUT);
    const float fca  = 0.5f * (__cosf(PI_F * d / ANG_CUT) + 1.0f);
    const float wa   = mang ? fca : 0.0f;   // lanes >= 24 carry 0 -> auto-mask

    s_u [wave][lane] = make_float4(ux, uy, uz, 0.0f);
    s_ws[wave][lane] = make_float2(wa, sj);
    // same-wave LDS ops complete in order; no barrier needed (no cross-wave sharing)

    // ---------------- Gram matrix U.U^T via V_WMMA_F32_16X16X4_F32 -----------
    // A (16x4 f32): lanes 0-15 hold (K0,K1)=(x,y) of row M=lane; lanes 16-31
    // hold (K2,K3)=(z,0) of row M=lane-16. B (4x16) packs identically.
    const int  lo = lane & 15;
    const bool hi = lane >= 16;

    const float4 ul = s_u[wave][lo];
    const float4 uh = s_u[wave][lo + 16];

    v2f a0, a1;
    a0.x = hi ? ul.z : ul.x;   a0.y = hi ? 0.0f : ul.y;   // neighbors 0..15
    a1.x = hi ? uh.z : uh.x;   a1.y = hi ? 0.0f : uh.y;   // neighbors 16..23 (+pad)

    v8f cz = {0.f, 0.f, 0.f, 0.f, 0.f, 0.f, 0.f, 0.f};
    v8f d00 = __builtin_amdgcn_wmma_f32_16x16x4_f32(false, a0, false, a0,
                                                    (short)0, cz, false, false);
    v8f d01 = __builtin_amdgcn_wmma_f32_16x16x4_f32(false, a0, false, a1,
                                                    (short)0, cz, false, false);
    v8f d11 = __builtin_amdgcn_wmma_f32_16x16x4_f32(false, a1, false, a1,
                                                    (short)0, cz, false, false);

    // ---------------- pair loop over held Gram entries ------------------------
    // D element (vgpr r, lane l): m = r + 8*(l/16), n = l%16.
    const int n = lo;
    const int h = lane >> 4;

    const float2 wk0 = s_ws[wave][n];        // k = n          (tile 00)
    const float2 wk1 = s_ws[wave][n + 16];   // k = 16+n, wa==0 if k>=24

    float au[9] = {0.f, 0.f, 0.f, 0.f, 0.f, 0.f, 0.f, 0.f, 0.f};
    float aw[9] = {0.f, 0.f, 0.f, 0.f, 0.f, 0.f, 0.f, 0.f, 0.f};

#pragma unroll
    for (int r = 0; r < 8; r++) {
        const int    jrow = r + 8 * h;              // 0..15
        const float2 wsl  = s_ws[wave][jrow];       // j-side, low block
        const float2 wsh  = s_ws[wave][jrow + 16];  // j-side, high block (0 if j>=24)
        const bool   lt   = jrow < n;               // j<k for tiles 00 and 11

        // tile 00: (j=jrow, k=n)
        const float w00 = lt ? wsl.x * wk0.x : 0.0f;
        cheb_acc(d00[r], w00, w00 * wsl.y * wk0.y, au, aw);
        // tile 01: (j=jrow, k=16+n) -- j<k always; k<24 enforced by wk1.x==0
        const float w01 = wsl.x * wk1.x;
        cheb_acc(d01[r], w01, w01 * wsl.y * wk1.y, au, aw);
        // tile 11: (j=16+jrow, k=16+n) -- j<24 enforced by wsh.x==0
        const float w11 = lt ? wsh.x * wk1.x : 0.0f;
        cheb_acc(d11[r], w11, w11 * wsh.y * wk1.y, au, aw);
    }

    // ---------------- angular reduction: one LDS transpose --------------------
    // rows 2i = au[i], rows 2i+1 = aw[i]; row stride RSTR dwords.
#pragma unroll
    for (int i = 0; i <= ANGO; i++) {
        s_red[wave][(2 * i)     * RSTR + lane] = au[i];
        s_red[wave][(2 * i + 1) * RSTR + lane] = aw[i];
    }
    const int row  = (lane < 18) ? lane : 0;
    const int base = row * RSTR;
    float sum = 0.0f;
#pragma unroll
    for (int l = 0; l < 32; l += 4) {
        const float4 q4 = *(const float4*)&s_red[wave][base + l];
        sum += (q4.x + q4.y) + (q4.z + q4.w);
    }
    if (lane < 18) {
        const int i = row >> 1;
        const int f = row & 1;                 // 0 = unweighted, 1 = spin-weighted
        out[atom * NOUT + 34 + f * 9 + i] = sum;
    }
}

extern "C" void kernel_launch(void* const* d_in, const int* in_sizes, int n_in,
                              void* d_out, int out_size, void* d_ws, size_t ws_size,
                              hipStream_t stream) {
    const float* pos     = (const float*)d_in[0];
    const int*   species = (const int*)d_in[1];
    const int*   nbr     = (const int*)d_in[2];
    float*       out     = (float*)d_out;

    dim3 grid(NATOMS / 8);   // 2500 blocks
    dim3 block(256);         // 8 wave32 per block, one atom per wave
    hipLaunchKernelGGL(cheb_desc_kernel, grid, block, 0, stream,
                       pos, species, nbr, out);
}